// LinkPredictionScorer_71648644432153
// MI455X (gfx1250) — compile-verified
//
#include <hip/hip_runtime.h>
#include <math.h>

// D = A(16x4 f32) * B(4x16 f32) + C(16x16 f32), wave32 WMMA
typedef __attribute__((ext_vector_type(2))) float v2f;
typedef __attribute__((ext_vector_type(8))) float v8f;

#define WAVES_PER_BLOCK 8
#define TILE_M 16
#define D_DIM 128
#define KSTEPS (D_DIM / 4)   // 32 WMMA steps
#define LDS_PITCH 132        // 128 + 4 pad: conflict-free strided b64 reads, 16B-aligned rows

// Phase 1: per-node scores s_src[n] = x[n].w_src, s_dst[n] = x[n].w_dst
// One 16-node tile per wave; K=128 swept in 32 WMMA(16x16x4) steps.
// B fragments (cols 0/1 = w_src/w_dst, rest 0) are pre-expanded into LDS in
// per-lane WMMA layout so the inner loop is ds_load_b64 x2 + v_wmma only.
__global__ __launch_bounds__(256)
void node_scores_wmma(const float* __restrict__ x,
                      const float* __restrict__ W,     // [2*128]: w_src | w_dst
                      float* __restrict__ sSrc,
                      float* __restrict__ sDst,
                      int nTiles)
{
    __shared__ float ldsX[WAVES_PER_BLOCK][TILE_M * LDS_PITCH]; // 66 KB
    __shared__ float bFrag[KSTEPS * 32 * 2];                    // 8 KB: [kstep][lane]{f2}

    const int tid  = threadIdx.x;
    const int lane = tid & 31;
    const int wave = tid >> 5;
    const int tile = blockIdx.x * WAVES_PER_BLOCK + wave;
    const bool valid = (tile < nTiles);

    // ---- Build B fragments once per block (1024 float2 entries, 4 per thread) ----
    #pragma unroll
    for (int j = 0; j < 4; ++j) {
        int e     = tid + j * 256;       // entry = kstep*32 + lane
        int ks    = e >> 5;
        int el    = e & 31;
        int em    = el & 15;             // N column
        int ehi   = el >> 4;             // K-half
        float bx = 0.0f, by = 0.0f;
        if (em < 2) {                    // col 0 = w_src, col 1 = w_dst
            const float* wp = W + em * D_DIM + ks * 4 + 2 * ehi;
            bx = wp[0];
            by = wp[1];
        }
        bFrag[e * 2 + 0] = bx;
        bFrag[e * 2 + 1] = by;
    }

    // ---- Stage one contiguous 16x128 fp32 tile of x into LDS (coalesced b128) ----
    if (valid) {
        const float4* src4 = (const float4*)(x + (size_t)tile * (TILE_M * D_DIM));
        #pragma unroll
        for (int i = 0; i < 16; ++i) {
            int f    = i * 32 + lane;    // float4 index within 8KB tile
            int row  = f >> 5;           // 32 float4 per 128-float row
            int col4 = f & 31;
            float4 v = src4[f];
            *(float4*)&ldsX[wave][row * LDS_PITCH + col4 * 4] = v;
        }
    }
    __syncthreads();
    if (!valid) return;

    const int m  = lane & 15;   // A: M row within tile; B/C/D: N column
    const int hi = lane >> 4;   // K-half selector for A fragments
    const float* aBase = &ldsX[wave][m * LDS_PITCH + 2 * hi];
    const float* bBase = &bFrag[lane * 2];

    v8f acc = {};
    #pragma unroll 8
    for (int ks = 0; ks < KSTEPS; ++ks) {
        v2f a = *(const v2f*)(aBase + ks * 4);
        v2f b = *(const v2f*)(bBase + ks * 64);
        // 8-arg form: (neg_a, A, neg_b, B, c_mod, C, reuse_a, reuse_b)
        acc = __builtin_amdgcn_wmma_f32_16x16x4_f32(false, a, false, b,
                                                    (short)0, acc, false, false);
    }

    // D column N: lanes N / N+16 hold M=0..7 / M=8..15 in acc[0..7]
    if (m < 2) {
        float* outp = (m == 0 ? sSrc : sDst) + (size_t)tile * TILE_M + hi * 8;
        float4 lo = { acc[0], acc[1], acc[2], acc[3] };
        float4 hv = { acc[4], acc[5], acc[6], acc[7] };
        *(float4*)(outp + 0) = lo;
        *(float4*)(outp + 4) = hv;
    }
}

// Phase 2: per-edge gather (L2-resident score arrays) + sigmoid, 4 edges/thread.
__global__ __launch_bounds__(256)
void edge_sigmoid(const float* __restrict__ sSrc,
                  const float* __restrict__ sDst,
                  const int* __restrict__ src,
                  const int* __restrict__ dst,
                  const float* __restrict__ b,
                  float* __restrict__ out,
                  int E)
{
    int q    = blockIdx.x * blockDim.x + threadIdx.x;
    int base = q * 4;
    if (base >= E) return;
    const float b0 = b[0];

    if (base + 3 < E) {
        int4 s = ((const int4*)src)[q];
        int4 d = ((const int4*)dst)[q];
        float4 r;
        r.x = 1.0f / (1.0f + __expf(-(sSrc[s.x] + sDst[d.x] + b0)));
        r.y = 1.0f / (1.0f + __expf(-(sSrc[s.y] + sDst[d.y] + b0)));
        r.z = 1.0f / (1.0f + __expf(-(sSrc[s.z] + sDst[d.z] + b0)));
        r.w = 1.0f / (1.0f + __expf(-(sSrc[s.w] + sDst[d.w] + b0)));
        ((float4*)out)[q] = r;
    } else {
        for (int e = base; e < E; ++e) {
            float z = sSrc[src[e]] + sDst[dst[e]] + b0;
            out[e] = 1.0f / (1.0f + __expf(-z));
        }
    }
}

extern "C" void kernel_launch(void* const* d_in, const int* in_sizes, int n_in,
                              void* d_out, int out_size, void* d_ws, size_t ws_size,
                              hipStream_t stream) {
    const float* x   = (const float*)d_in[0];   // (N,128) fp32
    const int*   src = (const int*)d_in[1];     // (E,)
    const int*   dst = (const int*)d_in[2];     // (E,)
    const float* W   = (const float*)d_in[3];   // (1,256) fp32
    const float* b   = (const float*)d_in[4];   // (1,)   fp32
    float* out = (float*)d_out;                 // (E,1)  fp32

    const int N = in_sizes[0] / D_DIM;          // 100000
    const int E = in_sizes[1];                  // 640000
    const int nTiles = N / TILE_M;              // 6250, exact

    float* sSrc = (float*)d_ws;                 // N floats
    float* sDst = sSrc + N;                     // N floats (800KB total, L2-resident)

    const int blocks1 = (nTiles + WAVES_PER_BLOCK - 1) / WAVES_PER_BLOCK;
    node_scores_wmma<<<blocks1, 256, 0, stream>>>(x, W, sSrc, sDst, nTiles);

    const int nQuads  = (E + 3) / 4;
    const int blocks2 = (nQuads + 255) / 256;
    edge_sigmoid<<<blocks2, 256, 0, stream>>>(sSrc, sDst, src, dst, b, out, E);
}